// Attention_local_40183714021382
// MI455X (gfx1250) — compile-verified
//
#include <hip/hip_runtime.h>
#include <hip/hip_bf16.h>

typedef __attribute__((ext_vector_type(2))) float v2f;
typedef __attribute__((ext_vector_type(8))) float v8f;

#define B_DIM 32
#define S_DIM 4096
#define D_DIM 512
#define WINW  50.0f
#define ROWS_PER_WG 64

// ---------------------------------------------------------------------------
// Zero the output accumulator (harness poisons d_out with 0xAA).
// ---------------------------------------------------------------------------
__global__ void zero_kernel(float* __restrict__ out, int n) {
    int i = blockIdx.x * blockDim.x + threadIdx.x;
    if (i < n) out[i] = 0.0f;
}

// ---------------------------------------------------------------------------
// Projections via fp32 WMMA 16x16x4:
//   which==0:  g = target @ Wa + ba            -> gvec (32 x 512)
//   which==1:  p = tanh(target @ Wp + bp)      -> pvec (32 x 512)
// Grid: (32 N-tiles, 2 M-tiles, 2 which), 1 wave (32 threads) per workgroup.
// A fragment (16x4 f32): lanes 0-15 hold K=k,k+1 ; lanes 16-31 hold K=k+2,k+3
// B fragment (4x16 f32): lane n in 0-15 holds col n for K=k,k+1 ;
//                        lane 16+n holds col n for K=k+2,k+3
// C (16x16 f32, 8 VGPRs): lane L -> col (L&15), rows = j + 8*(L>=16).
// ---------------------------------------------------------------------------
__global__ __launch_bounds__(32)
void proj_wmma_kernel(const float* __restrict__ target,
                      const float* __restrict__ Wa, const float* __restrict__ ba,
                      const float* __restrict__ Wp, const float* __restrict__ bp,
                      float* __restrict__ gvec, float* __restrict__ pvec)
{
    const int lane  = threadIdx.x;        // 0..31
    const int nTile = blockIdx.x;         // 0..31
    const int mTile = blockIdx.y;         // 0..1
    const int which = blockIdx.z;         // 0 -> Wa/gvec, 1 -> Wp/pvec

    const float* W    = which ? Wp : Wa;
    const float* bias = which ? bp : ba;
    float*       out  = which ? pvec : gvec;

    const int l15   = lane & 15;
    const int khalf = lane >> 4;          // 0 or 1
    const int arow  = mTile * 16 + l15;   // A row this lane supplies
    const int bcol  = nTile * 16 + l15;   // B col this lane supplies

    v8f acc = {};
    for (int k = 0; k < D_DIM; k += 4) {
        const int ka = k + 2 * khalf;
        v2f a, b;
        a.x = target[arow * D_DIM + ka];
        a.y = target[arow * D_DIM + ka + 1];
        b.x = W[(ka)     * D_DIM + bcol];
        b.y = W[(ka + 1) * D_DIM + bcol];
        acc = __builtin_amdgcn_wmma_f32_16x16x4_f32(
                  /*neg_a=*/false, a, /*neg_b=*/false, b,
                  /*c_mod=*/(short)0, acc, /*reuse_a=*/false, /*reuse_b=*/false);
    }

    const int cn    = nTile * 16 + l15;
    const int mbase = mTile * 16 + (khalf ? 8 : 0);
    const float bb  = bias[cn];
#pragma unroll
    for (int j = 0; j < 8; ++j) {
        float v = acc[j] + bb;
        if (which) v = tanhf(v);
        out[(mbase + j) * D_DIM + cn] = v;
    }
}

// ---------------------------------------------------------------------------
// aligned_position[b] = sigmoid(dot(pvec[b,:], Vp) + bv) * S
// One wave per batch row.
// ---------------------------------------------------------------------------
__global__ __launch_bounds__(32)
void align_kernel(const float* __restrict__ pvec, const float* __restrict__ Vp,
                  const float* __restrict__ bv, float* __restrict__ alignv)
{
    const int b    = blockIdx.x;
    const int lane = threadIdx.x;
    float s = 0.0f;
    for (int j = lane; j < D_DIM; j += 32)
        s += pvec[b * D_DIM + j] * Vp[j];
#pragma unroll
    for (int off = 16; off >= 1; off >>= 1)
        s += __shfl_xor(s, off, 32);
    if (lane == 0) {
        const float z   = s + bv[0];
        const float sig = 1.0f / (1.0f + __expf(-z));
        alignv[b] = sig * (float)S_DIM;
    }
}

// ---------------------------------------------------------------------------
// Fused double-softmax + gaussian window + weighted sum over S.
// Grid: (S/ROWS_PER_WG, B). Block = 256 threads = 8 waves; each wave owns
// whole 512-wide rows (16 f32/lane), wave-shuffle reductions, no barriers in
// the row loop. Chunks entirely outside the gaussian window are skipped:
// exp(-2*(d/50)^2) == 0.0f in f32 for |d| > ~360, so skipping is exact.
// ---------------------------------------------------------------------------
__global__ __launch_bounds__(256)
void context_kernel(const float* __restrict__ source,
                    const float* __restrict__ gvec,
                    const float* __restrict__ alignv,
                    float* __restrict__ outp)
{
    const int   b    = blockIdx.y;
    const int   s0   = blockIdx.x * ROWS_PER_WG;
    const float apos = alignv[b];

    // whole-chunk skip (uniform across workgroup)
    {
        const float dlo = (float)s0 - apos;
        const float dhi = (float)(s0 + ROWS_PER_WG - 1) - apos;
        float dmin = 0.0f;
        if (dlo > 0.0f)      dmin = dlo;
        else if (dhi < 0.0f) dmin = -dhi;
        if (dmin > 365.0f) return;   // gaussian underflows to exactly 0.0f
    }

    __shared__ float ctx[D_DIM];
    for (int i = threadIdx.x; i < D_DIM; i += 256) ctx[i] = 0.0f;
    __syncthreads();

    const int lane = threadIdx.x & 31;
    const int wave = threadIdx.x >> 5;

    // column mapping: cols(j) = (j>>2)*128 + lane*4 + (j&3), j = 0..15
    float gl[16];
    const float* gb = gvec + b * D_DIM;
#pragma unroll
    for (int q = 0; q < 4; ++q) {
        const float4 gv = *(const float4*)(gb + q * 128 + lane * 4);
        gl[q*4+0] = gv.x; gl[q*4+1] = gv.y; gl[q*4+2] = gv.z; gl[q*4+3] = gv.w;
    }

    float accl[16];
#pragma unroll
    for (int j = 0; j < 16; ++j) accl[j] = 0.0f;

    const float* srcB = source + (size_t)b * S_DIM * D_DIM;

    for (int s = s0 + wave; s < s0 + ROWS_PER_WG; s += 8) {
        const float d  = ((float)s - apos) * (1.0f / WINW);
        const float ex = -2.0f * d * d;
        if (ex < -100.0f) continue;           // gaussian == 0.0f exactly
        const float gf = __expf(ex);

        const float* srow = srcB + (size_t)s * D_DIM;
        float x[16];
#pragma unroll
        for (int q = 0; q < 4; ++q) {
            const float4 sv = *(const float4*)(srow + q * 128 + lane * 4);
            x[q*4+0] = sv.x; x[q*4+1] = sv.y; x[q*4+2] = sv.z; x[q*4+3] = sv.w;
        }

        // softmax #1 over D of (source * g)
        float t[16];
        float m = -__builtin_inff();
#pragma unroll
        for (int j = 0; j < 16; ++j) { t[j] = x[j] * gl[j]; m = fmaxf(m, t[j]); }
#pragma unroll
        for (int off = 16; off >= 1; off >>= 1) m = fmaxf(m, __shfl_xor(m, off, 32));
        float sum = 0.0f;
#pragma unroll
        for (int j = 0; j < 16; ++j) { t[j] = __expf(t[j] - m); sum += t[j]; }
#pragma unroll
        for (int off = 16; off >= 1; off >>= 1) sum += __shfl_xor(sum, off, 32);
        const float inv = 1.0f / sum;

        // softmax #2 over D of softmax #1
        float u[16];
        float m2 = -__builtin_inff();
#pragma unroll
        for (int j = 0; j < 16; ++j) { u[j] = t[j] * inv; m2 = fmaxf(m2, u[j]); }
#pragma unroll
        for (int off = 16; off >= 1; off >>= 1) m2 = fmaxf(m2, __shfl_xor(m2, off, 32));
        float sum2 = 0.0f;
#pragma unroll
        for (int j = 0; j < 16; ++j) { u[j] = __expf(u[j] - m2); sum2 += u[j]; }
#pragma unroll
        for (int off = 16; off >= 1; off >>= 1) sum2 += __shfl_xor(sum2, off, 32);

        const float w = gf / sum2;
#pragma unroll
        for (int j = 0; j < 16; ++j) accl[j] += u[j] * w * x[j];
    }

    // combine waves in LDS (ds_add_f32), then one global atomic pass
#pragma unroll
    for (int q = 0; q < 4; ++q)
#pragma unroll
        for (int r = 0; r < 4; ++r)
            atomicAdd(&ctx[q * 128 + lane * 4 + r], accl[q * 4 + r]);
    __syncthreads();

    for (int i = threadIdx.x; i < D_DIM; i += 256)
        atomicAdd(&outp[b * D_DIM + i], ctx[i]);
}

// ---------------------------------------------------------------------------
// Launch
// ---------------------------------------------------------------------------
extern "C" void kernel_launch(void* const* d_in, const int* in_sizes, int n_in,
                              void* d_out, int out_size, void* d_ws, size_t ws_size,
                              hipStream_t stream) {
    const float* source = (const float*)d_in[0];
    const float* target = (const float*)d_in[1];
    const float* Wp     = (const float*)d_in[2];
    const float* bp     = (const float*)d_in[3];
    const float* Wa     = (const float*)d_in[4];
    const float* ba     = (const float*)d_in[5];
    const float* Vp     = (const float*)d_in[6];
    const float* bvv    = (const float*)d_in[7];

    float* out  = (float*)d_out;
    float* ws   = (float*)d_ws;
    float* gvec   = ws;                       // 32*512 floats
    float* pvec   = ws + B_DIM * D_DIM;       // 32*512 floats
    float* alignv = ws + 2 * B_DIM * D_DIM;   // 32 floats

    zero_kernel<<<(B_DIM * D_DIM + 255) / 256, 256, 0, stream>>>(out, B_DIM * D_DIM);

    dim3 g1(32, 2, 2);
    proj_wmma_kernel<<<g1, 32, 0, stream>>>(target, Wa, ba, Wp, bp, gvec, pvec);

    align_kernel<<<B_DIM, 32, 0, stream>>>(pvec, Vp, bvv, alignv);

    dim3 g2(S_DIM / ROWS_PER_WG, B_DIM);
    context_kernel<<<g2, 256, 0, stream>>>(source, gvec, alignv, out);
}